// MultiMonotoneHollowConv_70300024701179
// MI455X (gfx1250) — compile-verified
//
#include <hip/hip_runtime.h>
#include <hip/hip_bf16.h>

typedef __attribute__((ext_vector_type(2))) float v2f;
typedef __attribute__((ext_vector_type(8))) float v8f;

// Only check on the DEVICE pass: __has_builtin() reports false for amdgcn
// target builtins during the host (x86) pass even though calls are legal.
#if defined(__HIP_DEVICE_COMPILE__)
#if !__has_builtin(__builtin_amdgcn_wmma_f32_16x16x4_f32)
#error "wmma_f32_16x16x4_f32 builtin not available on device toolchain"
#endif
#endif

__device__ __forceinline__ v8f wmma_f32_4(v2f a, v2f b, v8f c) {
  // 8-arg pattern: (neg_a, A, neg_b, B, c_mod, C, reuse_a, reuse_b)
  return __builtin_amdgcn_wmma_f32_16x16x4_f32(false, a, false, b, (short)0, c,
                                               false, false);
}

// ---------------------------------------------------------------------------
// Forward conv via implicit GEMM on V_WMMA_F32_16X16X4_F32.
//   y[b][oc][oh][ow] = sum_{ic,kh,kw} w[oc][ic][kh][kw] *
//                      x[b][ic][oh*s - p + kh][ow*s - p + kw]
// M = 16 output channels per tile (1 or 2 tiles, compile-time), N = 16 output
// pixels, K = IC*9 stepped by 4. One wave (32 threads) per workgroup.
// All shape/stride parameters are template constants so the accumulators are
// plain named registers (no dynamically indexed VGPR arrays -> no v_movrel).
// ---------------------------------------------------------------------------
template <int IC, int OC, int STRIDE, int PAD, int ACCUM>
__global__ __launch_bounds__(32) void conv_fwd_wmma(
    const float* __restrict__ x, const float* __restrict__ w,
    float* __restrict__ y, int B, int H, int W, int OH, int OW) {
  constexpr int CS = 15 * STRIDE + 3;  // staged tile colspan
  constexpr int K = IC * 9;
  __shared__ float xs[IC * 3 * CS];

  const int ntw = (OW + 15) >> 4;
  int bid = blockIdx.x;
  const int wt = bid % ntw; bid /= ntw;
  const int oh = bid % OH;  bid /= OH;
  const int b  = bid;
  const int lane = threadIdx.x;
  const int w0 = wt << 4;

  const int base_r = oh * STRIDE - PAD;
  const int base_c = w0 * STRIDE - PAD;

  // Stage input halo tile (zero-padded) into LDS.
  for (int i = lane; i < IC * 3 * CS; i += 32) {
    const int ic = i / (3 * CS);
    const int rem = i - ic * 3 * CS;
    const int rr = rem / CS, cc = rem - rr * CS;
    const int gr = base_r + rr, gc = base_c + cc;
    float v = 0.f;
    if (gr >= 0 && gr < H && gc >= 0 && gc < W)
      v = x[((size_t)(b * IC + ic) * H + gr) * W + gc];
    xs[i] = v;
  }
  __syncthreads();

  v8f acc0 = {};
  v8f acc1 = {};
  const int n  = lane & 15;
  const int m  = lane & 15;
  const int hi = lane >> 4;  // lane half

  for (int k0 = 0; k0 < K; k0 += 4) {
    // B fragment: VGPR v holds row K=k0+v (lanes 0-15) / K=k0+v+2 (16-31).
    const int kb0 = k0 + (hi ? 2 : 0);
    const int kb1 = kb0 + 1;
    const int ic0 = kb0 / 9, r0 = kb0 % 9;
    const int ic1 = kb1 / 9, r1 = kb1 % 9;
    v2f bf;
    bf.x = xs[(ic0 * 3 + r0 / 3) * CS + n * STRIDE + (r0 % 3)];
    bf.y = xs[(ic1 * 3 + r1 / 3) * CS + n * STRIDE + (r1 % 3)];

    // A fragment: VGPR v holds K=k0+v (lanes 0-15) / K=k0+v+2 (16-31).
    {
      const size_t wrow = (size_t)m * (IC * 9);
      v2f af;
      af.x = w[wrow + kb0];
      af.y = w[wrow + kb1];
      acc0 = wmma_f32_4(af, bf, acc0);
    }
    if (OC > 16) {
      const size_t wrow = (size_t)(16 + m) * (IC * 9);
      v2f af;
      af.x = w[wrow + kb0];
      af.y = w[wrow + kb1];
      acc1 = wmma_f32_4(af, bf, acc1);
    }
  }

  // Store D: VGPR v -> M = v + 8*(lane>=16), N = lane&15.
  const int mbase = hi * 8;
  if (w0 + n < OW) {
#pragma unroll
    for (int v = 0; v < 8; ++v) {
      const int oc = mbase + v;
      const size_t idx = ((size_t)(b * OC + oc) * OH + oh) * OW + (w0 + n);
      float out = acc0[v];
      if (ACCUM) out += y[idx];
      y[idx] = out;
    }
    if (OC > 16) {
#pragma unroll
      for (int v = 0; v < 8; ++v) {
        const int oc = 16 + mbase + v;
        const size_t idx = ((size_t)(b * OC + oc) * OH + oh) * OW + (w0 + n);
        float out = acc1[v];
        if (ACCUM) out += y[idx];
        y[idx] = out;
      }
    }
  }
}

// ---------------------------------------------------------------------------
// Transpose conv via implicit GEMM.
//   z[b][m][oh][ow] (+)= sum_{o,kh,kw} yin[b][o][(oh+p-kh)/s][(ow+p-kw)/s]
//                         * w[o][m][kh][kw]      (terms valid iff divisible)
// M = 16 output channels per tile (1 or 2 tiles, compile-time), N = 16 output
// pixels, K = OCin*9 stepped by 4. Stride-2 uses parity-masked B values.
// ---------------------------------------------------------------------------
template <int OCIN, int MC, int STRIDE, int PAD, int ACCUM>
__global__ __launch_bounds__(32) void conv_t_wmma(
    const float* __restrict__ yin, const float* __restrict__ w,
    float* __restrict__ z, int B, int IH, int IW, int OH, int OW) {
  constexpr int CS = (STRIDE == 1) ? 18 : 11;  // staged tile colspan
  constexpr int K = OCIN * 9;
  __shared__ float ys[OCIN * 3 * CS];

  const int ntw = OW >> 4;
  int bid = blockIdx.x;
  const int wt = bid % ntw; bid /= ntw;
  const int oh = bid % OH;  bid /= OH;
  const int b  = bid;
  const int lane = threadIdx.x;
  const int w0 = wt << 4;

  int rbase, cbase;
  if (STRIDE == 1) { rbase = oh + PAD - 2; cbase = w0 + PAD - 2; }
  else             { rbase = (oh + 1) >> 1; cbase = (w0 + 1) >> 1; }

  for (int i = lane; i < OCIN * 3 * CS; i += 32) {
    const int o = i / (3 * CS);
    const int rem = i - o * 3 * CS;
    const int rr = rem / CS, cc = rem - rr * CS;
    const int gr = rbase + rr, gc = cbase + cc;
    float v = 0.f;
    if (gr >= 0 && gr < IH && gc >= 0 && gc < IW)
      v = yin[((size_t)(b * OCIN + o) * IH + gr) * IW + gc];
    ys[i] = v;
  }
  __syncthreads();

  v8f acc0 = {};
  v8f acc1 = {};
  const int n  = lane & 15;
  const int m  = lane & 15;
  const int hi = lane >> 4;

  for (int k0 = 0; k0 < K; k0 += 4) {
    const int kb0 = k0 + (hi ? 2 : 0);
    const int kb1 = kb0 + 1;

    v2f bf;
#pragma unroll
    for (int v = 0; v < 2; ++v) {
      const int k = (v == 0) ? kb0 : kb1;
      const int o = k / 9, r9 = k % 9, kh = r9 / 3, kw = r9 % 3;
      float val;
      if (STRIDE == 1) {
        val = ys[(o * 3 + (2 - kh)) * CS + (n + 2 - kw)];
      } else {
        const int t = oh + PAD - kh;
        const int u = w0 + n + PAD - kw;
        if ((t & 1) | (u & 1)) {
          val = 0.f;
        } else {
          val = ys[(o * 3 + ((t >> 1) - rbase)) * CS + ((u >> 1) - cbase)];
        }
      }
      if (v == 0) bf.x = val; else bf.y = val;
    }

    {
      const int o0 = kb0 / 9, r90 = kb0 % 9;
      const int o1 = kb1 / 9, r91 = kb1 % 9;
      v2f af;
      af.x = w[(size_t)(o0 * MC + m) * 9 + r90];
      af.y = w[(size_t)(o1 * MC + m) * 9 + r91];
      acc0 = wmma_f32_4(af, bf, acc0);
      if (MC > 16) {
        v2f ag;
        ag.x = w[(size_t)(o0 * MC + 16 + m) * 9 + r90];
        ag.y = w[(size_t)(o1 * MC + 16 + m) * 9 + r91];
        acc1 = wmma_f32_4(ag, bf, acc1);
      }
    }
  }

  const int mbase = hi * 8;
#pragma unroll
  for (int v = 0; v < 8; ++v) {
    const int mm = mbase + v;
    const size_t idx = ((size_t)(b * MC + mm) * OH + oh) * OW + (w0 + n);
    float out = acc0[v];
    if (ACCUM) out += z[idx];
    z[idx] = out;
  }
  if (MC > 16) {
#pragma unroll
    for (int v = 0; v < 8; ++v) {
      const int mm = 16 + mbase + v;
      const size_t idx = ((size_t)(b * MC + mm) * OH + oh) * OW + (w0 + n);
      float out = acc1[v];
      if (ACCUM) out += z[idx];
      z[idx] = out;
    }
  }
}

// ---------------------------------------------------------------------------
// Prep: build blk0 (2,2,16,16) and blk1 (2,16,16) self-term matrices.
// ---------------------------------------------------------------------------
__global__ void prep_kernel(const float* __restrict__ W00,
                            const float* __restrict__ W10,
                            const float* __restrict__ W11,
                            float* __restrict__ blk0, float* __restrict__ blk1) {
  const int t = threadIdx.x;
  if (blockIdx.x == 0) {
    if (t < 256) {
      const int m = t >> 4, nn = t & 15;
      float sall = 0.f;
      float rc[2][2] = {{0.f, 0.f}, {0.f, 0.f}};
      for (int i1 = 0; i1 < 3; ++i1)
        for (int j1 = 0; j1 < 3; ++j1) {
          float s00 = 0.f, s10 = 0.f;
          for (int o = 0; o < 16; ++o)
            s00 += W00[((o * 16 + m) * 3 + i1) * 3 + j1] *
                   W00[((o * 16 + nn) * 3 + i1) * 3 + j1];
          for (int o = 0; o < 32; ++o)
            s10 += W10[((o * 16 + m) * 3 + i1) * 3 + j1] *
                   W10[((o * 16 + nn) * 3 + i1) * 3 + j1];
          sall += s00;                       // stride 1: hits every (r,c)
          rc[(i1 + 1) & 1][(j1 + 1) & 1] += s10;  // stride 2, offset 1
        }
      for (int r = 0; r < 2; ++r)
        for (int c = 0; c < 2; ++c)
          blk0[((r * 2 + c) * 16 + m) * 16 + nn] = sall + rc[r][c];
    }
  } else {
    if (t < 512) {
      const int g = t >> 8, mn = t & 255, m = mn >> 4, nn = mn & 15;
      float s = 0.f;
      for (int i1 = 0; i1 < 3; ++i1)
        for (int j1 = 0; j1 < 3; ++j1)
          for (int o = 0; o < 32; ++o)
            s += W11[((o * 32 + g * 16 + m) * 3 + i1) * 3 + j1] *
                 W11[((o * 32 + g * 16 + nn) * 3 + i1) * 3 + j1];
      blk1[(g * 16 + m) * 16 + nn] = s;
    }
  }
}

// Deterministic two-pass ||b||^2 reduction (no float atomics).
__global__ void norm_partial(const float* __restrict__ b0,
                             const float* __restrict__ b1,
                             float* __restrict__ partial, long n0, long n1) {
  __shared__ float s0s[256], s1s[256];
  float s0 = 0.f, s1 = 0.f;
  const long stridelen = (long)gridDim.x * blockDim.x;
  for (long i = (long)blockIdx.x * blockDim.x + threadIdx.x; i < n0;
       i += stridelen) { const float v = b0[i]; s0 += v * v; }
  for (long i = (long)blockIdx.x * blockDim.x + threadIdx.x; i < n1;
       i += stridelen) { const float v = b1[i]; s1 += v * v; }
  s0s[threadIdx.x] = s0; s1s[threadIdx.x] = s1;
  __syncthreads();
  for (int off = 128; off > 0; off >>= 1) {
    if ((int)threadIdx.x < off) {
      s0s[threadIdx.x] += s0s[threadIdx.x + off];
      s1s[threadIdx.x] += s1s[threadIdx.x + off];
    }
    __syncthreads();
  }
  if (threadIdx.x == 0) {
    partial[blockIdx.x * 2 + 0] = s0s[0];
    partial[blockIdx.x * 2 + 1] = s1s[0];
  }
}

__global__ void norm_finish(const float* __restrict__ partial,
                            float* __restrict__ norms) {
  __shared__ float a[512], c[512];
  const int t = threadIdx.x;
  a[t] = partial[2 * t]; c[t] = partial[2 * t + 1];
  __syncthreads();
  for (int off = 256; off > 0; off >>= 1) {
    if (t < off) { a[t] += a[t + off]; c[t] += c[t + off]; }
    __syncthreads();
  }
  if (t == 0) { norms[0] = a[0]; norms[1] = c[0]; }
}

// out0 = -(Z0 - blk0-diag(x0) - b0_hat*g0), in place on d_out.
__global__ void epilogue0(float* __restrict__ z, const float* __restrict__ x0,
                          const float* __restrict__ bias,
                          const float* __restrict__ blk,
                          const float* __restrict__ norms,
                          const float* __restrict__ g, long total) {
  __shared__ float sb[1024];
  for (int i = threadIdx.x; i < 1024; i += blockDim.x) sb[i] = blk[i];
  __syncthreads();
  const float inv = g[0] * rsqrtf(norms[0]);
  const long stridelen = (long)gridDim.x * blockDim.x;
  for (long idx = (long)blockIdx.x * blockDim.x + threadIdx.x; idx < total;
       idx += stridelen) {
    const int wv = (int)(idx & 255); long r = idx >> 8;
    const int hv = (int)(r & 255);  r >>= 8;
    const int m = (int)(r & 15);
    const int b = (int)(r >> 4);
    const float* xp = x0 + (((long)b * 16) * 256 + hv) * 256 + wv;
    const float* bl = sb + ((((hv & 1) * 2 + (wv & 1)) * 16 + m) * 16);
    float diag = 0.f;
#pragma unroll
    for (int nn = 0; nn < 16; ++nn) diag += bl[nn] * xp[(long)nn * 65536];
    const float bt = bias[((long)m * 256 + hv) * 256 + wv] * inv;
    z[idx] = diag + bt - z[idx];
  }
}

// out1 = -(Z1 - blk1-diag(x1) - b1_hat*g1), in place on d_out.
__global__ void epilogue1(float* __restrict__ z, const float* __restrict__ x1,
                          const float* __restrict__ bias,
                          const float* __restrict__ blk,
                          const float* __restrict__ norms,
                          const float* __restrict__ g, long total) {
  __shared__ float sb[512];
  for (int i = threadIdx.x; i < 512; i += blockDim.x) sb[i] = blk[i];
  __syncthreads();
  const float inv = g[1] * rsqrtf(norms[1]);
  const long stridelen = (long)gridDim.x * blockDim.x;
  for (long idx = (long)blockIdx.x * blockDim.x + threadIdx.x; idx < total;
       idx += stridelen) {
    const int wv = (int)(idx & 127); long r = idx >> 7;
    const int hv = (int)(r & 127);   r >>= 7;
    const int c = (int)(r & 31);
    const int b = (int)(r >> 5);
    const int gg = c >> 4, m = c & 15;
    const float* xp = x1 + (((long)(b * 32 + gg * 16)) * 128 + hv) * 128 + wv;
    const float* bl = sb + (gg * 16 + m) * 16;
    float diag = 0.f;
#pragma unroll
    for (int nn = 0; nn < 16; ++nn) diag += bl[nn] * xp[(long)nn * 16384];
    const float bt = bias[((long)c * 128 + hv) * 128 + wv] * inv;
    z[idx] = diag + bt - z[idx];
  }
}

extern "C" void kernel_launch(void* const* d_in, const int* in_sizes, int n_in,
                              void* d_out, int out_size, void* d_ws,
                              size_t ws_size, hipStream_t stream) {
  const float* x0  = (const float*)d_in[0];  // (16,16,256,256)
  const float* x1  = (const float*)d_in[1];  // (16,32,128,128)
  const float* W00 = (const float*)d_in[2];  // (16,16,3,3)
  const float* W10 = (const float*)d_in[3];  // (32,16,3,3)
  const float* W11 = (const float*)d_in[4];  // (32,32,3,3)
  const float* b0  = (const float*)d_in[5];  // (16,256,256)
  const float* b1  = (const float*)d_in[6];  // (32,128,128)
  const float* g   = (const float*)d_in[7];  // (2,)

  float* ws      = (float*)d_ws;
  float* norms   = ws;                 // 2
  float* blk0    = ws + 16;            // 1024
  float* blk1    = ws + 16 + 1024;     // 512
  float* partial = ws + 1600;          // 1024
  float* Y0      = ws + 4096;                        // 16*16*258*258
  float* Y1      = Y0 + (size_t)16 * 16 * 258 * 258; // 16*32*130*130

  float* out0 = (float*)d_out;
  float* out1 = out0 + (size_t)16 * 16 * 256 * 256;

  // Self-term matrices + deterministic bias norms.
  prep_kernel<<<2, 512, 0, stream>>>(W00, W10, W11, blk0, blk1);
  norm_partial<<<512, 256, 0, stream>>>(b0, b1, partial,
                                        (long)16 * 256 * 256,
                                        (long)32 * 128 * 128);
  norm_finish<<<1, 512, 0, stream>>>(partial, norms);

  // Forward: Y0 = conv(x0,W00,s1,p2) -> (16,16,258,258)
  conv_fwd_wmma<16, 16, 1, 2, 0><<<dim3(16 * 258 * 17), 32, 0, stream>>>(
      x0, W00, Y0, 16, 256, 256, 258, 258);
  // Y1 = conv(x0,W10,s2,p3) + conv(x1,W11,s1,p2) -> (16,32,130,130)
  conv_fwd_wmma<16, 32, 2, 3, 0><<<dim3(16 * 130 * 9), 32, 0, stream>>>(
      x0, W10, Y1, 16, 256, 256, 130, 130);
  conv_fwd_wmma<32, 32, 1, 2, 1><<<dim3(16 * 130 * 9), 32, 0, stream>>>(
      x1, W11, Y1, 16, 128, 128, 130, 130);

  // Transpose: Z0 = conv_t(Y0,W00,1,2,0) + conv_t(Y1,W10,2,3,1) -> out0
  conv_t_wmma<16, 16, 1, 2, 0><<<dim3(16 * 256 * 16), 32, 0, stream>>>(
      Y0, W00, out0, 16, 258, 258, 256, 256);
  conv_t_wmma<32, 16, 2, 3, 1><<<dim3(16 * 256 * 16), 32, 0, stream>>>(
      Y1, W10, out0, 16, 130, 130, 256, 256);
  // Z1 = conv_t(Y1,W11,1,2,0) -> out1
  conv_t_wmma<32, 32, 1, 2, 0><<<dim3(16 * 128 * 8), 32, 0, stream>>>(
      Y1, W11, out1, 16, 130, 130, 128, 128);

  // Epilogues: subtract block-diag self-term + normalized bias, negate.
  epilogue0<<<4096, 256, 0, stream>>>(out0, x0, b0, blk0, norms, g,
                                      (long)16 * 16 * 256 * 256);
  epilogue1<<<4096, 256, 0, stream>>>(out1, x1, b1, blk1, norms, g,
                                      (long)16 * 32 * 128 * 128);
}